// Net_81707457839588
// MI455X (gfx1250) — compile-verified
//
#include <hip/hip_runtime.h>
#include <hip/hip_bf16.h>

#define NQ      10
#define LAYERS  10
#define BATCH   4096
#define NACT    4

typedef __attribute__((ext_vector_type(2))) float v2f;
typedef __attribute__((ext_vector_type(8))) float v8f;

// ---------------------------------------------------------------------------
// Setup: build per-layer Kronecker factors
//   A_l = Kron(R_{l,0..4})  (32x32 complex, acts on row index / qubits 0-4)
//   B_l = Kron(R_{l,5..9})  (32x32 complex, acts on col index / qubits 5-9)
// ws layout per layer l, per factor (floats):
//   [real 1024][imag 1024][neg-imag 1024]   (A then B -> 6144 floats/layer)
// The negated-imag plane lets the complex GEMM run with zero runtime VALU
// negations (F32 WMMA has no A/B-negate modifier).
// ---------------------------------------------------------------------------
__global__ void build_gates_kernel(const float* __restrict__ w,
                                   float* __restrict__ ws) {
  int id = blockIdx.x * blockDim.x + threadIdx.x;
  if (id >= LAYERS * 2 * 1024) return;
  int l     = id >> 11;
  int rem   = id & 2047;
  int which = rem >> 10;       // 0 = A (qubits 0-4), 1 = B (qubits 5-9)
  int e     = rem & 1023;
  int r  = e >> 5;
  int rp = e & 31;
  float accr = 1.f, acci = 0.f;
  for (int j = 0; j < 5; ++j) {
    int q = which * 5 + j;
    float phi   = w[(l * NQ + q) * 3 + 0];
    float theta = w[(l * NQ + q) * 3 + 1];
    float omega = w[(l * NQ + q) * 3 + 2];
    float ct = cosf(0.5f * theta), st = sinf(0.5f * theta);
    float a = 0.5f * (phi + omega), b = 0.5f * (phi - omega);
    float ca = cosf(a), sa = sinf(a), cb = cosf(b), sb = sinf(b);
    // Rot = [[ep*ct, -em*st], [conj(em)*st, conj(ep)*ct]]
    //   ep = exp(-i a) = (ca, -sa);  em = exp(+i b) = (cb, sb)
    int bi = (r  >> (4 - j)) & 1;
    int bj = (rp >> (4 - j)) & 1;
    float gr, gi;
    if (!bi && !bj)      { gr =  ca * ct; gi = -sa * ct; }
    else if (!bi &&  bj) { gr = -cb * st; gi = -sb * st; }
    else if ( bi && !bj) { gr =  cb * st; gi = -sb * st; }
    else                 { gr =  ca * ct; gi =  sa * ct; }
    float nr = accr * gr - acci * gi;
    float ni = accr * gi + acci * gr;
    accr = nr; acci = ni;
  }
  float* base = ws + l * 6144 + which * 3072;
  base[e]        =  accr;   // real plane
  base[1024 + e] =  acci;   // imag plane
  base[2048 + e] = -acci;   // negated imag plane
}

// Apply the layer-l CNOT ring (r = l%9+1) scatter map to index b.
__device__ __forceinline__ int cnot_map(int b, int rr) {
  #pragma unroll
  for (int w = 0; w < NQ; ++w) {
    int t = w + rr; if (t >= NQ) t -= NQ;
    b ^= ((b >> (9 - w)) & 1) << (9 - t);
  }
  return b;
}

// ---------------------------------------------------------------------------
// Main kernel: one wave32 per batch item. State lives in this wave's LDS
// slice as Sr/Si (32x32, linear index = full 10-bit basis index).
// Per layer:  T = S * B^T  (WMMA, accum in VGPRs, store to T buffer)
//             S = A * T    (WMMA), written back with the CNOT-ring
//                          permutation fused into the LDS addresses,
//                          using GF(2)-linearity of the composed map.
// ---------------------------------------------------------------------------
__global__ __launch_bounds__(64)
void qsim_kernel(const float* __restrict__ x,
                 const float* __restrict__ gates,
                 const float* __restrict__ fcw,
                 const float* __restrict__ fcb,
                 float* __restrict__ out) {
  __shared__ float lds[2 * 4096];            // 2 waves * (Sr,Si,Tr,Ti)
  const int wave = threadIdx.x >> 5;
  const int lane = threadIdx.x & 31;
  const int half = lane >> 4;                // which 16-lane half
  const int l16  = lane & 15;
  const int b    = blockIdx.x * 2 + wave;

  float* Sr = lds + wave * 4096;
  float* Si = Sr + 1024;
  float* Tr = Sr + 2048;
  float* Ti = Sr + 3072;

  // ---- input layer: relu + RX angles; closed-form initial state ----------
  float cw[NQ], sw[NQ];
  for (int w = 0; w < NQ; ++w) {
    float xv = x[b * NQ + w];
    xv = xv > 0.f ? xv : 0.f;
    cw[w] = cosf(0.5f * xv);
    sw[w] = sinf(0.5f * xv);
  }
  {
    // amplitude(idx) = (prod_w bit?sin:cos) * (-i)^popcount(idx)
    float mhi = 1.f;
    for (int w = 0; w < 5; ++w) mhi *= ((lane >> (4 - w)) & 1) ? sw[w] : cw[w];
    int pchi = __popc(lane);
    for (int c = 0; c < 32; ++c) {
      float m = mhi;
      for (int w = 0; w < 5; ++w) m *= ((c >> (4 - w)) & 1) ? sw[5 + w] : cw[5 + w];
      int k = (pchi + __popc(c)) & 3;
      float re = 0.f, im = 0.f;
      if      (k == 0) re =  m;
      else if (k == 1) im = -m;
      else if (k == 2) re = -m;
      else             im =  m;
      Sr[lane * 32 + c] = re;
      Si[lane * 32 + c] = im;
    }
  }
  __syncthreads();

  // ---- variational layers -------------------------------------------------
  for (int l = 0; l < LAYERS; ++l) {
    const float* gAr  = gates + l * 6144;
    const float* gAi  = gAr + 1024;
    const float* gnAi = gAr + 2048;
    const float* gBr  = gAr + 3072;
    const float* gBi  = gAr + 4096;
    const float* gnBi = gAr + 5120;

    // CNOT-ring permutation, exploited as GF(2)-linear map:
    //   P(idx) = XOR of images of idx's set bits.
    const int rr = l % 9 + 1;
    const int pI = cnot_map(1 << 9, rr);     // image of tile-row bit  (I)
    const int p7 = cnot_map(1 << 7, rr);     // images of v bits
    const int p6 = cnot_map(1 << 6, rr);
    const int p5 = cnot_map(1 << 5, rr);
    const int pJ = cnot_map(1 << 4, rr);     // image of tile-col bit  (J)
    // lane-fixed part: column l16 plus the half bit (idx bit 8)
    const int pLane = cnot_map((half << 8) | l16, rr);

    // Step 1: T = S @ B^T   (complex; -Bi plane replaces runtime negation)
    for (int I = 0; I < 2; ++I)
    for (int J = 0; J < 2; ++J) {
      v8f Rr = {}, Ri = {};
      const int row = I * 16 + l16;          // A-operand row (per lane)
      const int n   = J * 16 + l16;          // B-operand col (per lane)
      for (int K = 0; K < 2; ++K)
      for (int kk = 0; kk < 4; ++kk) {
        const int k0 = K * 16 + kk * 4 + 2 * half;
        v2f sr   = *(const v2f*)(Sr   + row * 32 + k0);
        v2f si   = *(const v2f*)(Si   + row * 32 + k0);
        v2f btr  = *(const v2f*)(gBr  + n * 32 + k0);   // B^T[k][n] = B[n][k]
        v2f bti  = *(const v2f*)(gBi  + n * 32 + k0);
        v2f nbti = *(const v2f*)(gnBi + n * 32 + k0);
        Rr = __builtin_amdgcn_wmma_f32_16x16x4_f32(false, sr, false, btr,  (short)0, Rr, false, false);
        Rr = __builtin_amdgcn_wmma_f32_16x16x4_f32(false, si, false, nbti, (short)0, Rr, false, false);
        Ri = __builtin_amdgcn_wmma_f32_16x16x4_f32(false, sr, false, bti,  (short)0, Ri, false, false);
        Ri = __builtin_amdgcn_wmma_f32_16x16x4_f32(false, si, false, btr,  (short)0, Ri, false, false);
      }
      #pragma unroll
      for (int v = 0; v < 8; ++v) {          // C/D layout: m = v + 8*half
        int m = I * 16 + v + 8 * half;
        Tr[m * 32 + n] = Rr[v];
        Ti[m * 32 + n] = Ri[v];
      }
    }
    __syncthreads();

    // Step 2: S' = A @ T, written back through the CNOT permutation
    for (int I = 0; I < 2; ++I)
    for (int J = 0; J < 2; ++J) {
      v8f Ur = {}, Ui = {};
      const int row = I * 16 + l16;
      const int n   = J * 16 + l16;
      for (int K = 0; K < 2; ++K)
      for (int kk = 0; kk < 4; ++kk) {
        const int k0 = K * 16 + kk * 4 + 2 * half;
        v2f ar  = *(const v2f*)(gAr  + row * 32 + k0);
        v2f ai  = *(const v2f*)(gAi  + row * 32 + k0);
        v2f nai = *(const v2f*)(gnAi + row * 32 + k0);
        v2f btr, bti;                         // B-op[k][n] = T[k][n], k = k0+v
        btr.x = Tr[(k0 + 0) * 32 + n];
        btr.y = Tr[(k0 + 1) * 32 + n];
        bti.x = Ti[(k0 + 0) * 32 + n];
        bti.y = Ti[(k0 + 1) * 32 + n];
        Ur = __builtin_amdgcn_wmma_f32_16x16x4_f32(false, ar,  false, btr, (short)0, Ur, false, false);
        Ur = __builtin_amdgcn_wmma_f32_16x16x4_f32(false, nai, false, bti, (short)0, Ur, false, false);
        Ui = __builtin_amdgcn_wmma_f32_16x16x4_f32(false, ar,  false, bti, (short)0, Ui, false, false);
        Ui = __builtin_amdgcn_wmma_f32_16x16x4_f32(false, ai,  false, btr, (short)0, Ui, false, false);
      }
      // destination base for this tile (constant-folded XOR combine)
      const int pbase = pLane ^ (I ? pI : 0) ^ (J ? pJ : 0);
      #pragma unroll
      for (int v = 0; v < 8; ++v) {
        int pb = pbase ^ ((v & 4) ? p7 : 0) ^ ((v & 2) ? p6 : 0) ^ ((v & 1) ? p5 : 0);
        Sr[pb] = Ur[v];
        Si[pb] = Ui[v];
      }
    }
    __syncthreads();
  }

  // ---- readout: z_w = sum |amp|^2 * (1 - 2*bit_w), then 10->4 FC ----------
  float z[NQ];
  for (int w = 0; w < NQ; ++w) z[w] = 0.f;
  for (int c = 0; c < 32; ++c) {
    int idx = lane * 32 + c;
    float re = Sr[idx], im = Si[idx];
    float p = re * re + im * im;
    #pragma unroll
    for (int w = 0; w < NQ; ++w)
      z[w] += ((idx >> (9 - w)) & 1) ? -p : p;
  }
  for (int off = 16; off > 0; off >>= 1)
    #pragma unroll
    for (int w = 0; w < NQ; ++w)
      z[w] += __shfl_xor(z[w], off, 32);
  if (lane < NACT) {
    float acc = fcb[lane];
    for (int w = 0; w < NQ; ++w) acc += fcw[lane * NQ + w] * z[w];
    out[b * NACT + lane] = acc;
  }
}

// ---------------------------------------------------------------------------
extern "C" void kernel_launch(void* const* d_in, const int* in_sizes, int n_in,
                              void* d_out, int out_size, void* d_ws, size_t ws_size,
                              hipStream_t stream) {
  const float* inputs  = (const float*)d_in[0];  // (4096, 10)
  const float* weights = (const float*)d_in[1];  // (10, 10, 3)
  const float* fcw     = (const float*)d_in[2];  // (4, 10)
  const float* fcb     = (const float*)d_in[3];  // (4,)
  float* out   = (float*)d_out;                  // (4096, 4)
  float* gates = (float*)d_ws;                   // needs 10*6144*4 = 240 KB

  build_gates_kernel<<<80, 256, 0, stream>>>(weights, gates);
  qsim_kernel<<<BATCH / 2, 64, 0, stream>>>(inputs, gates, fcw, fcb, out);
}